// Pool_88776974008963
// MI455X (gfx1250) — compile-verified
//
#include <hip/hip_runtime.h>
#include <hip/hip_bf16.h>

typedef __attribute__((ext_vector_type(2))) float v2f;
typedef __attribute__((ext_vector_type(8))) float v8f;

#define B_SZ   64
#define T_SZ   2048
#define D_SZ   768
#define P_SZ   1024
#define K_TOP  8
#define TSPLIT 16
#define TROWS  (T_SZ / TSPLIT)   // 128
#define ROWV4  (D_SZ / 4)        // 192 float4 per row

// ---------------- workspace layout (float offsets) ----------------
#define OFF_XMEAN  ((size_t)0)                       // 64*768   = 49152
#define OFF_SIM    ((size_t)49152)                   // 64*1024  = 65536
#define OFF_XPROJ  ((size_t)114688)                  // 49152
#define OFF_XNORM  ((size_t)163840)                  // 49152
#define OFF_COUNTS ((size_t)212992)                  // 1024 ints
#define OFF_MAJOR  ((size_t)214016)                  // 8 ints (+pad)
#define OFF_PNORM  ((size_t)214528)                  // 1024*768 = 786432
#define OFF_PART   OFF_PNORM                         // partials overlap prompt_norm (disjoint lifetimes)

// ---------------- kernel 1: partial max over token chunks ----------------
// grid (TSPLIT, B), block = 192 threads; one float4 column per thread.
__global__ void max_partial_k(const float* __restrict__ x, float* __restrict__ part) {
    const int b   = blockIdx.y;
    const int tc  = blockIdx.x;
    const int tid = threadIdx.x;              // 0..191
    const float4* xp = (const float4*)(x) + ((size_t)b * T_SZ + (size_t)tc * TROWS) * ROWV4 + tid;
    float4 m = xp[0];
    #pragma unroll 4
    for (int t = 1; t < TROWS; ++t) {
        float4 v = xp[(size_t)t * ROWV4];
        m.x = fmaxf(m.x, v.x); m.y = fmaxf(m.y, v.y);
        m.z = fmaxf(m.z, v.z); m.w = fmaxf(m.w, v.w);
    }
    float4* pp = (float4*)(part) + ((size_t)tc * B_SZ + b) * ROWV4 + tid;
    *pp = m;
}

// ---------------- kernel 2: finalize max + zero counts ----------------
__global__ void max_finalize_k(const float* __restrict__ part, float* __restrict__ x_mean,
                               int* __restrict__ counts) {
    const int i = blockIdx.x * blockDim.x + threadIdx.x;
    if (i < B_SZ * D_SZ) {
        float m = part[i];
        #pragma unroll
        for (int c = 1; c < TSPLIT; ++c) m = fmaxf(m, part[(size_t)c * B_SZ * D_SZ + i]);
        x_mean[i] = m;
    }
    if (i < P_SZ) counts[i] = 0;
}

// ---------------- kernel 3: row-wise l2 normalize (width = 768) ----------------
__global__ void l2norm_rows_k(const float* __restrict__ in, float* __restrict__ out) {
    __shared__ float red[256];
    const int row = blockIdx.x;
    const float* ip = in + (size_t)row * D_SZ;
    float s = 0.0f;
    for (int d = threadIdx.x; d < D_SZ; d += 256) { float v = ip[d]; s += v * v; }
    red[threadIdx.x] = s; __syncthreads();
    for (int off = 128; off > 0; off >>= 1) {
        if (threadIdx.x < off) red[threadIdx.x] += red[threadIdx.x + off];
        __syncthreads();
    }
    const float scale = 1.0f / sqrtf(fmaxf(red[0], 1e-12f));
    for (int d = threadIdx.x; d < D_SZ; d += 256) out[(size_t)row * D_SZ + d] = ip[d] * scale;
}

// ---------------- WMMA f32 GEMM: C[M,N] = A[M,K] * Brows[N,K]^T ----------------
// One wave (32 threads) per 16x16 tile. grid = (M/16, N/16), block = 32.
__global__ void gemm_nt_wmma_k(const float* __restrict__ A, const float* __restrict__ Brows,
                               float* __restrict__ C, int N, int K) {
    const int lane = threadIdx.x;
    const int half = lane >> 4;        // 0 or 1
    const int l16  = lane & 15;
    const int arow = blockIdx.x * 16 + l16;     // M index
    const int brow = blockIdx.y * 16 + l16;     // N index (Brows is [N][K])
    const float* Ap = A     + (size_t)arow * K + 2 * half;
    const float* Bp = Brows + (size_t)brow * K + 2 * half;
    v8f acc = {};
    #pragma unroll 4
    for (int kk = 0; kk < K; kk += 4) {
        v2f a, b;
        a.x = Ap[kk]; a.y = Ap[kk + 1];     // A frag: M=lane%16, K=kk + v + 2*(lane/16)
        b.x = Bp[kk]; b.y = Bp[kk + 1];     // B frag mirrored: N=lane%16
        acc = __builtin_amdgcn_wmma_f32_16x16x4_f32(false, a, false, b,
                                                    (short)0, acc, false, false);
    }
    // D layout: N = lane%16, M = v + 8*(lane/16)
    float* Cp = C + ((size_t)blockIdx.x * 16 + 8 * half) * N + blockIdx.y * 16 + l16;
    #pragma unroll
    for (int v = 0; v < 8; ++v) Cp[(size_t)v * N] = acc[v];
}

// ---------------- kernel 6: per-row top-8 -> bincount (atomic int add) ----------------
__global__ void topk_rows_k(const float* __restrict__ sim, int* __restrict__ counts) {
    __shared__ float vals[P_SZ];
    __shared__ float rv[256];
    __shared__ int   ri[256];
    __shared__ int   sel[K_TOP];
    const int row = blockIdx.x;
    const float* sp = sim + (size_t)row * P_SZ;
    for (int i = threadIdx.x; i < P_SZ; i += 256) vals[i] = sp[i];
    __syncthreads();
    for (int pass = 0; pass < K_TOP; ++pass) {
        float bv = -__builtin_inff(); int bi = 0x7fffffff;
        for (int i = threadIdx.x; i < P_SZ; i += 256) {
            float v = vals[i];
            if (v > bv) { bv = v; bi = i; }      // ascending i => first occurrence kept on ties
        }
        rv[threadIdx.x] = bv; ri[threadIdx.x] = bi;
        __syncthreads();
        for (int off = 128; off > 0; off >>= 1) {
            if (threadIdx.x < off) {
                float ov = rv[threadIdx.x + off]; int oi = ri[threadIdx.x + off];
                if (ov > rv[threadIdx.x] || (ov == rv[threadIdx.x] && oi < ri[threadIdx.x])) {
                    rv[threadIdx.x] = ov; ri[threadIdx.x] = oi;
                }
            }
            __syncthreads();
        }
        if (threadIdx.x == 0) { sel[pass] = ri[0]; vals[ri[0]] = -__builtin_inff(); }
        __syncthreads();
    }
    if (threadIdx.x < K_TOP) atomicAdd(&counts[sel[threadIdx.x]], 1);
}

// ---------------- kernel 7: top-8 over counts (desc count, asc id) ----------------
__global__ void topk_counts_k(const int* __restrict__ counts, int* __restrict__ major) {
    __shared__ int cv[P_SZ];
    __shared__ int rv[256];
    __shared__ int ri[256];
    for (int i = threadIdx.x; i < P_SZ; i += 256) cv[i] = counts[i];
    __syncthreads();
    for (int pass = 0; pass < K_TOP; ++pass) {
        int bv = -1, bi = 0x7fffffff;
        for (int i = threadIdx.x; i < P_SZ; i += 256) {
            int v = cv[i];
            if (v > bv) { bv = v; bi = i; }
        }
        rv[threadIdx.x] = bv; ri[threadIdx.x] = bi;
        __syncthreads();
        for (int off = 128; off > 0; off >>= 1) {
            if (threadIdx.x < off) {
                int ov = rv[threadIdx.x + off], oi = ri[threadIdx.x + off];
                if (ov > rv[threadIdx.x] || (ov == rv[threadIdx.x] && oi < ri[threadIdx.x])) {
                    rv[threadIdx.x] = ov; ri[threadIdx.x] = oi;
                }
            }
            __syncthreads();
        }
        if (threadIdx.x == 0) { major[pass] = ri[0]; cv[ri[0]] = -1; }
        __syncthreads();
    }
}

// ---------------- kernel 8: batched_prompt gather -> out[1..] ----------------
__global__ void gather_out_k(const float* __restrict__ prompt, const int* __restrict__ major,
                             float* __restrict__ out) {
    const int i = blockIdx.x * 256 + threadIdx.x;   // over 64*8*768
    if (i >= B_SZ * K_TOP * D_SZ) return;
    const int d = i % D_SZ;
    const int j = (i / D_SZ) % K_TOP;
    out[1 + i] = prompt[(size_t)major[j] * D_SZ + d];
}

// ---------------- kernel 9: reduce_sim scalar -> out[0] ----------------
// sum_{b,j,d} pn[mid_j][d]*x_norm[b][d] / B  ==  dot(sum_j pn[mid_j], sum_b x_norm) / B
__global__ void reduce_sim_k(const float* __restrict__ prompt_norm, const float* __restrict__ x_norm,
                             const int* __restrict__ major, float* __restrict__ out) {
    __shared__ float red[256];
    float acc = 0.0f;
    for (int d = threadIdx.x; d < D_SZ; d += 256) {
        float xs = 0.0f;
        for (int b = 0; b < B_SZ; ++b) xs += x_norm[(size_t)b * D_SZ + d];
        float ps = 0.0f;
        #pragma unroll
        for (int j = 0; j < K_TOP; ++j) ps += prompt_norm[(size_t)major[j] * D_SZ + d];
        acc += xs * ps;
    }
    red[threadIdx.x] = acc; __syncthreads();
    for (int off = 128; off > 0; off >>= 1) {
        if (threadIdx.x < off) red[threadIdx.x] += red[threadIdx.x + off];
        __syncthreads();
    }
    if (threadIdx.x == 0) out[0] = red[0] / (float)B_SZ;
}

extern "C" void kernel_launch(void* const* d_in, const int* in_sizes, int n_in,
                              void* d_out, int out_size, void* d_ws, size_t ws_size,
                              hipStream_t stream) {
    const float* x_embed = (const float*)d_in[0];   // [64, 2048, 768]
    const float* prompt  = (const float*)d_in[1];   // [1024, 768]
    const float* W       = (const float*)d_in[2];   // [768, 768]
    float* out = (float*)d_out;                     // [1 + 64*8*768]
    float* ws  = (float*)d_ws;

    float* x_mean      = ws + OFF_XMEAN;
    float* sim         = ws + OFF_SIM;
    float* x_proj      = ws + OFF_XPROJ;
    float* x_norm      = ws + OFF_XNORM;
    int*   counts      = (int*)(ws + OFF_COUNTS);
    int*   major       = (int*)(ws + OFF_MAJOR);
    float* prompt_norm = ws + OFF_PNORM;
    float* partials    = ws + OFF_PART;             // overlaps prompt_norm (disjoint lifetime)

    // 1) token-max reduction (bandwidth-dominant pass: 403 MB)
    max_partial_k<<<dim3(TSPLIT, B_SZ), 192, 0, stream>>>(x_embed, partials);
    max_finalize_k<<<(B_SZ * D_SZ + 255) / 256, 256, 0, stream>>>(partials, x_mean, counts);

    // 2) prompt_norm
    l2norm_rows_k<<<P_SZ, 256, 0, stream>>>(prompt, prompt_norm);

    // 3) x_proj = x_mean @ W^T   (WMMA f32, 4x48 tiles)
    gemm_nt_wmma_k<<<dim3(B_SZ / 16, D_SZ / 16), 32, 0, stream>>>(x_mean, W, x_proj, D_SZ, D_SZ);

    // 4) x_norm
    l2norm_rows_k<<<B_SZ, 256, 0, stream>>>(x_proj, x_norm);

    // 5) sim = x_norm @ prompt_norm^T  (WMMA f32, 4x64 tiles)
    gemm_nt_wmma_k<<<dim3(B_SZ / 16, P_SZ / 16), 32, 0, stream>>>(x_norm, prompt_norm, sim, P_SZ, D_SZ);

    // 6) per-row top-8 -> counts ; 7) global top-8 ids
    topk_rows_k<<<B_SZ, 256, 0, stream>>>(sim, counts);
    topk_counts_k<<<1, 256, 0, stream>>>(counts, major);

    // 8) outputs
    gather_out_k<<<(B_SZ * K_TOP * D_SZ + 255) / 256, 256, 0, stream>>>(prompt, major, out);
    reduce_sim_k<<<1, 256, 0, stream>>>(prompt_norm, x_norm, major, out);
}